// Encoder_5531917878006
// MI455X (gfx1250) — compile-verified
//
#include <hip/hip_runtime.h>
#include <math.h>

typedef __attribute__((ext_vector_type(2))) float v2f;
typedef __attribute__((ext_vector_type(4))) float v4f;
typedef __attribute__((ext_vector_type(8))) float v8f;

#define D1 768
#define SEQ 512
#define NROW (8 * SEQ)      // 4096 (b,s) rows
#define WREL_ROWS 130       // 2*MAXOFF + 2
#define EPS 1e-12f

// ---------------------------------------------------------------------------
// Generic: LayerNorm over 64-wide rows; optional broadcast over `nbcast`
// copies spaced `bstride` floats apart. Used for embd0 (W_glob, x8 batches)
// and for precomputing ln(W_rel) rows into workspace.
// ---------------------------------------------------------------------------
__global__ void ln64_kernel(const float* __restrict__ in, float* __restrict__ out,
                            int nbcast, int bstride) {
  __shared__ float buf[64];
  __shared__ float st[2];
  const int r = blockIdx.x, t = threadIdx.x;
  const float v = in[r * 64 + t];
  buf[t] = v;
  __syncthreads();
  if (t == 0) {
    float s = 0.f;
    for (int k = 0; k < 64; ++k) s += buf[k];
    const float mean = s * (1.0f / 64.0f);
    float q = 0.f;
    for (int k = 0; k < 64; ++k) { const float d = buf[k] - mean; q += d * d; }
    st[0] = mean;
    st[1] = 1.0f / sqrtf(q * (1.0f / 64.0f) + EPS);
  }
  __syncthreads();
  const float o = (v - st[0]) * st[1];
  for (int bb = 0; bb < nbcast; ++bb)
    out[(size_t)bb * bstride + r * 64 + t] = o;
}

// ---------------------------------------------------------------------------
// embd1: raw = W_word[tok] + W_type[tt] + W_abs[s]  (to workspace, feeds GEMM)
//        out1 = LayerNorm(raw) over D1=768. One 256-thread block per row.
// ---------------------------------------------------------------------------
__global__ __launch_bounds__(256) void embd1_kernel(
    const int* __restrict__ tok, const int* __restrict__ tty,
    const float* __restrict__ Ww, const float* __restrict__ Wt,
    const float* __restrict__ Wa, float* __restrict__ out1,
    float* __restrict__ raw) {
  __shared__ float s1[256], s2[256];
  const int row = blockIdx.x, t = threadIdx.x;
  const int s = row & (SEQ - 1);
  const int tk = tok[row], ty = tty[row];
  const float* w0 = Ww + (size_t)tk * D1;
  const float* w1 = Wt + (size_t)ty * D1;
  const float* w2 = Wa + (size_t)s * D1;
  const float x0 = w0[t]       + w1[t]       + w2[t];
  const float x1 = w0[t + 256] + w1[t + 256] + w2[t + 256];
  const float x2 = w0[t + 512] + w1[t + 512] + w2[t + 512];
  float* rr = raw + (size_t)row * D1;
  rr[t] = x0; rr[t + 256] = x1; rr[t + 512] = x2;
  s1[t] = x0 + x1 + x2;
  s2[t] = x0 * x0 + x1 * x1 + x2 * x2;
  __syncthreads();
  for (int off = 128; off > 0; off >>= 1) {
    if (t < off) { s1[t] += s1[t + off]; s2[t] += s2[t + off]; }
    __syncthreads();
  }
  const float mean = s1[0] * (1.0f / 768.0f);
  const float var  = s2[0] * (1.0f / 768.0f) - mean * mean;
  const float rstd = 1.0f / sqrtf(var + EPS);
  float* oo = out1 + (size_t)row * D1;
  oo[t]       = (x0 - mean) * rstd;
  oo[t + 256] = (x1 - mean) * rstd;
  oo[t + 512] = (x2 - mean) * rstd;
}

// ---------------------------------------------------------------------------
// diag2[4096,64] = raw[4096,768] @ W_diag^T[768,64] + b_diag, full f32 via
// V_WMMA_F32_16X16X4_F32. One wave32 per 16x16 output tile, 192 K-steps.
// ISA f32 layouts (§7.12.2): A 16x4: lane holds row M=lane%16, lane-half picks
// K pair {0,1}/{2,3} in 2 VGPRs. B 4x16: lane holds col N=lane%16, same K
// split. C/D: VGPR r -> row M = r + 8*(lane>>4), col N = lane%16.
// ---------------------------------------------------------------------------
__global__ __launch_bounds__(256) void diag_wmma_kernel(
    const float* __restrict__ raw, const float* __restrict__ Wd,
    const float* __restrict__ bd, float* __restrict__ diag2) {
  const int wave = threadIdx.x >> 5, lane = threadIdx.x & 31;
  const int tile = blockIdx.x * 8 + wave;          // 1024 tiles = 256 x 4
  const int m0 = (tile >> 2) * 16, n0 = (tile & 3) * 16;
  const int l16 = lane & 15, half = lane >> 4;
  const float* arow = raw + (size_t)(m0 + l16) * D1 + half * 2;  // A row, K-major
  const float* brow = Wd  + (size_t)(n0 + l16) * D1 + half * 2;  // W_diag[n][k]
  v8f acc = {};
#pragma unroll 4
  for (int k = 0; k < D1; k += 4) {
    const v2f a = *(const v2f*)(arow + k);
    const v2f b = *(const v2f*)(brow + k);
    acc = __builtin_amdgcn_wmma_f32_16x16x4_f32(
        /*neg_a=*/false, a, /*neg_b=*/false, b,
        /*c_mod=*/(short)0, acc, /*reuse_a=*/false, /*reuse_b=*/false);
  }
  const float bias = bd[n0 + l16];
  const int mbase = m0 + half * 8;
#pragma unroll
  for (int r = 0; r < 8; ++r)
    diag2[(size_t)(mbase + r) * 64 + n0 + l16] = acc[r] + bias;
}

// ---------------------------------------------------------------------------
// embd2: the 536 MB streaming kernel. One block per (b,i) row-plane.
// Bulk: out[b,i,j,:] = lds_ln_wrel[rpe(b,i,j)] (b128 LDS read -> NT b128 store)
// Diagonal: LN(W_rel[0] + diag2[b,i]) written separately (bulk skips j==i).
// ---------------------------------------------------------------------------
__global__ __launch_bounds__(256) void embd2_kernel(
    const int* __restrict__ tty, const float* __restrict__ Wrel,
    const float* __restrict__ lnw, const float* __restrict__ diag2,
    float* __restrict__ out2) {
  __shared__ __align__(16) float s_lnw[WREL_ROWS * 64];  // 33 KB
  __shared__ int s_type[SEQ];
  __shared__ float s_diag[64];
  __shared__ float s_st[2];
  const int t = threadIdx.x;
  const int b = blockIdx.x >> 9, i = blockIdx.x & 511;

  for (int idx = t; idx < WREL_ROWS * 16; idx += 256)
    ((v4f*)s_lnw)[idx] = ((const v4f*)lnw)[idx];
  for (int idx = t; idx < SEQ; idx += 256)
    s_type[idx] = tty[b * SEQ + idx];
  __syncthreads();

  const int tt_i = s_type[i];
  const int jj = t >> 4, d4 = t & 15;          // 16 j's x 16 float4-chunks
  const size_t rowbase = (size_t)(b * SEQ + i) * SEQ;

  for (int jb = 0; jb < SEQ; jb += 16) {
    const int j = jb + jj;
    const int dd = i - j;
    int v;
    if (dd == 0)      v = 0;
    else if (dd > 0)  v = 128 - (dd < 63 ? dd : 63);
    else            { const int nd = -dd; v = (nd < 63 ? nd : 63); }
    if (s_type[j] != tt_i) v = 64;             // cross-segment -> MAXOFF
    if (i == 0 && j > 0)   v = 128;            // row-0 override
    if (j == 0 && i > 0)   v = 129;            // col-0 override
    if (j != i) {
      const v4f val = ((const v4f*)(s_lnw + v * 64))[d4];
      __builtin_nontemporal_store(val, (v4f*)(out2 + (rowbase + j) * 64) + d4);
    }
  }

  // diagonal: LN(W_rel[0] + diag2[b,i]) over 64
  if (t < 64) s_diag[t] = Wrel[t] + diag2[(size_t)(b * SEQ + i) * 64 + t];
  __syncthreads();
  if (t == 0) {
    float s = 0.f;
    for (int k = 0; k < 64; ++k) s += s_diag[k];
    const float mean = s * (1.0f / 64.0f);
    float q = 0.f;
    for (int k = 0; k < 64; ++k) { const float d = s_diag[k] - mean; q += d * d; }
    s_st[0] = mean;
    s_st[1] = 1.0f / sqrtf(q * (1.0f / 64.0f) + EPS);
  }
  __syncthreads();
  if (t < 64)
    out2[(rowbase + i) * 64 + t] = (s_diag[t] - s_st[0]) * s_st[1];
}

// ---------------------------------------------------------------------------
extern "C" void kernel_launch(void* const* d_in, const int* in_sizes, int n_in,
                              void* d_out, int out_size, void* d_ws, size_t ws_size,
                              hipStream_t stream) {
  (void)in_sizes; (void)n_in; (void)out_size; (void)ws_size;
  const int*   tok   = (const int*)d_in[0];
  const int*   tty   = (const int*)d_in[1];
  const float* Wword = (const float*)d_in[2];
  const float* Wtype = (const float*)d_in[3];
  const float* Wabs  = (const float*)d_in[4];
  const float* Wrel  = (const float*)d_in[5];
  const float* Wglob = (const float*)d_in[6];
  const float* Wdiag = (const float*)d_in[7];
  const float* bdiag = (const float*)d_in[8];

  float* out  = (float*)d_out;
  float* out0 = out;                       // [8,64,64]
  float* out1 = out + 32768;               // [8,512,768]
  float* out2 = out + 32768 + 3145728;     // [8,512,512,64]

  float* wsf    = (float*)d_ws;
  float* lnw    = wsf;                     // 130*64          = 8320 f
  float* raw    = wsf + 8320;              // 4096*768        = 3145728 f
  float* diag2w = wsf + 8320 + 3145728;    // 4096*64         = 262144 f (~13.7 MB total)

  ln64_kernel<<<64, 64, 0, stream>>>(Wglob, out0, 8, 4096);        // embd0
  ln64_kernel<<<WREL_ROWS, 64, 0, stream>>>(Wrel, lnw, 1, 0);      // ln(W_rel) table
  embd1_kernel<<<NROW, 256, 0, stream>>>(tok, tty, Wword, Wtype, Wabs, out1, raw);
  diag_wmma_kernel<<<128, 256, 0, stream>>>(raw, Wdiag, bdiag, diag2w);
  embd2_kernel<<<NROW, 256, 0, stream>>>(tty, Wrel, lnw, diag2w, out2);
}